// DGConv_41137196761670
// MI455X (gfx1250) — compile-verified
//
#include <hip/hip_runtime.h>

#define D 64

typedef float v2f __attribute__((ext_vector_type(2)));
typedef float v8f __attribute__((ext_vector_type(8)));

// deg[i] = 1 (self loop)
__global__ void dg_deg_init(float* __restrict__ deg, int n) {
    int i = blockIdx.x * blockDim.x + threadIdx.x;
    if (i < n) deg[i] = 1.0f;
}

// deg[dst[e]] += 1 for every real edge
__global__ void dg_deg_accum(const int* __restrict__ dst, float* __restrict__ deg, int e) {
    int i = blockIdx.x * blockDim.x + threadIdx.x;
    if (i < e) atomicAdd(&deg[dst[i]], 1.0f);
}

// deg -> d^{-1/2} in place
__global__ void dg_deg_to_dinv(float* __restrict__ deg, int n) {
    int i = blockIdx.x * blockDim.x + threadIdx.x;
    if (i < n) {
        float d = deg[i];
        deg[i] = (d > 0.0f) ? rsqrtf(d) : 0.0f;
    }
}

__global__ void dg_zero_f4(float4* __restrict__ p, int n4) {
    int i = blockIdx.x * blockDim.x + threadIdx.x;
    if (i < n4) p[i] = make_float4(0.f, 0.f, 0.f, 0.f);
}

// One wave32 per edge: lanes cover 64 features as 32 x float2.
// acc[dst] += dinv[src]*dinv[dst] * x[src]
__global__ void dg_scatter(const float* __restrict__ x,
                           const int* __restrict__ src,
                           const int* __restrict__ dst,
                           const float* __restrict__ dinv,
                           float* __restrict__ acc, int e) {
    int tid  = blockIdx.x * blockDim.x + threadIdx.x;
    int eidx = tid >> 5;
    int lane = tid & 31;
    if (eidx >= e) return;
    int s = src[eidx];
    int d = dst[eidx];
    float w = dinv[s] * dinv[d];
    float2 v = ((const float2*)(x + (size_t)s * D))[lane];
    float* ad = acc + (size_t)d * D + lane * 2;
    atomicAdd(ad + 0, w * v.x);
    atomicAdd(ad + 1, w * v.y);
}

// x_out = (1-delta)*x_in + delta*(acc + dinv[i]^2 * x_in)   (self-loop folded in)
__global__ void dg_update(const float* __restrict__ x_in,
                          const float* __restrict__ acc,
                          const float* __restrict__ dinv,
                          float* __restrict__ x_out, int n, float delta) {
    int tid  = blockIdx.x * blockDim.x + threadIdx.x;
    int i    = tid >> 5;
    int lane = tid & 31;
    if (i >= n) return;
    float di = dinv[i];
    float ws = di * di;
    float2 xv = ((const float2*)(x_in + (size_t)i * D))[lane];
    float2 av = ((const float2*)(acc + (size_t)i * D))[lane];
    float2 o;
    o.x = (1.0f - delta) * xv.x + delta * (av.x + ws * xv.x);
    o.y = (1.0f - delta) * xv.y + delta * (av.y + ws * xv.y);
    ((float2*)(x_out + (size_t)i * D))[lane] = o;
}

// out[16-row tile, 0:64] = X[tile] @ W + b using V_WMMA_F32_16X16X4_F32.
// One wave per tile; four 16x16 C accumulators cover the 64 output columns.
// Layouts per CDNA5 ISA 7.12.2:
//   A 16x4 f32: lane l -> row (l&15); VGPR0/1 hold K {0,1} (lanes 0-15) or {2,3} (lanes 16-31)
//   B 4x16 f32: lane l -> col (l&15); same half-split over K
//   C/D 16x16 : lane l -> col (l&15); VGPR v -> row v + 8*(l>=16)
__global__ void dg_gemm_wmma(const float* __restrict__ X,
                             const float* __restrict__ Wm,
                             const float* __restrict__ bias,
                             float* __restrict__ out, int n) {
    int tile  = blockIdx.x;
    int lane  = threadIdx.x;           // 0..31, full wave, no divergence around WMMA
    int m     = lane & 15;
    int khalf = (lane >> 4) * 2;       // 0 or 2
    int row0  = tile * 16;

    v8f c0 = {}, c1 = {}, c2 = {}, c3 = {};

    int rowA = row0 + m;
    if (rowA > n - 1) rowA = n - 1;    // clamp loads; EXEC stays all-ones
    const float* xrow = X + (size_t)rowA * D;

    #pragma unroll
    for (int k0 = 0; k0 < D; k0 += 4) {
        v2f a;
        a.x = xrow[k0 + khalf + 0];
        a.y = xrow[k0 + khalf + 1];

        const float* w0 = Wm + (size_t)(k0 + khalf) * D;   // row k
        const float* w1 = w0 + D;                          // row k+1

        v2f b0; b0.x = w0[ 0 + m]; b0.y = w1[ 0 + m];
        v2f b1; b1.x = w0[16 + m]; b1.y = w1[16 + m];
        v2f b2; b2.x = w0[32 + m]; b2.y = w1[32 + m];
        v2f b3; b3.x = w0[48 + m]; b3.y = w1[48 + m];

        c0 = __builtin_amdgcn_wmma_f32_16x16x4_f32(false, a, false, b0, (short)0, c0, false, false);
        c1 = __builtin_amdgcn_wmma_f32_16x16x4_f32(false, a, false, b1, (short)0, c1, false, false);
        c2 = __builtin_amdgcn_wmma_f32_16x16x4_f32(false, a, false, b2, (short)0, c2, false, false);
        c3 = __builtin_amdgcn_wmma_f32_16x16x4_f32(false, a, false, b3, (short)0, c3, false, false);
    }

    float bb0 = bias[ 0 + m];
    float bb1 = bias[16 + m];
    float bb2 = bias[32 + m];
    float bb3 = bias[48 + m];
    int rbase = (lane >> 4) * 8;
    #pragma unroll
    for (int v = 0; v < 8; ++v) {
        int rr = row0 + rbase + v;
        if (rr < n) {
            float* orow = out + (size_t)rr * D;
            orow[ 0 + m] = c0[v] + bb0;
            orow[16 + m] = c1[v] + bb1;
            orow[32 + m] = c2[v] + bb2;
            orow[48 + m] = c3[v] + bb3;
        }
    }
}

extern "C" void kernel_launch(void* const* d_in, const int* in_sizes, int n_in,
                              void* d_out, int out_size, void* d_ws, size_t ws_size,
                              hipStream_t stream) {
    const float* x    = (const float*)d_in[0];
    const int*   ei   = (const int*)d_in[1];
    const float* Wm   = (const float*)d_in[2];
    const float* bias = (const float*)d_in[3];
    float* out = (float*)d_out;

    const int n = in_sizes[0] / D;       // 100000
    const int e = in_sizes[1] / 2;       // 1600000
    const int* src = ei;
    const int* dst = ei + e;

    // workspace: dinv (n f32) | xbuf (n*64 f32) | acc (n*64 f32)
    char* ws = (char*)d_ws;
    float* dinv = (float*)ws;
    size_t off = ((size_t)n * sizeof(float) + 255) & ~(size_t)255;
    float* xbuf = (float*)(ws + off);
    off += (((size_t)n * D * sizeof(float)) + 255) & ~(size_t)255;
    float* acc = (float*)(ws + off);

    const float delta = 5.27f / 2.0f;    // T / K

    const int B = 256;
    dg_deg_init   <<<(n + B - 1) / B, B, 0, stream>>>(dinv, n);
    dg_deg_accum  <<<(e + B - 1) / B, B, 0, stream>>>(dst, dinv, e);
    dg_deg_to_dinv<<<(n + B - 1) / B, B, 0, stream>>>(dinv, n);

    const int nf4 = n * D / 4;
    const long scThreads = (long)e * 32;
    const int  scBlocks  = (int)((scThreads + B - 1) / B);
    const long upThreads = (long)n * 32;
    const int  upBlocks  = (int)((upThreads + B - 1) / B);

    // step 1: reads input x (never mutated), writes xbuf
    dg_zero_f4<<<(nf4 + B - 1) / B, B, 0, stream>>>((float4*)acc, nf4);
    dg_scatter<<<scBlocks, B, 0, stream>>>(x, src, dst, dinv, acc, e);
    dg_update <<<upBlocks, B, 0, stream>>>(x, acc, dinv, xbuf, n, delta);

    // step 2: xbuf -> xbuf (in-place safe: each thread rewrites only its own elems)
    dg_zero_f4<<<(nf4 + B - 1) / B, B, 0, stream>>>((float4*)acc, nf4);
    dg_scatter<<<scBlocks, B, 0, stream>>>(xbuf, src, dst, dinv, acc, e);
    dg_update <<<upBlocks, B, 0, stream>>>(xbuf, acc, dinv, xbuf, n, delta);

    // projection: out = xbuf @ W + b  (fp32 WMMA, one wave per 16-row tile)
    const int tiles = (n + 15) / 16;
    dg_gemm_wmma<<<tiles, 32, 0, stream>>>(xbuf, Wm, bias, out, n);
}